// AggregateJoint_8435315769722
// MI455X (gfx1250) — compile-verified
//
#include <hip/hip_runtime.h>

// All GEMMs via V_WMMA_F32_16X16X4_F32 (fp32 in/out, matches reference math).
typedef float v2f __attribute__((ext_vector_type(2)));
typedef float v8f __attribute__((ext_vector_type(8)));

constexpr int P_ = 16, IN_ = 4, H_ = 16, O_ = 3;
constexpr int BATCH = 256, NCOL = 64, FEAT = 512;
constexpr int ROWS  = BATCH * FEAT;      // 131072 rows of xr
constexpr int TPW   = 4;                 // 16-row WMMA tiles per wave
constexpr int WROWS = 16 * TPW;          // 64 rows (features) per wave
constexpr int WAVES = 8;                 // waves per block (256 threads)
constexpr int JOBS  = ROWS / WROWS;      // 2048 wave-jobs
constexpr int BLOCKS = JOBS / WAVES;     // 256 blocks
constexpr float SLOPE = 0.01f, EPS = 1e-5f;

// Block LDS layout (float offsets):
constexpr int LW1 = 0;                   // W1 copy           [16][4][16]  = 1024
constexpr int LW2 = 1024;                // W2 zero-padded    [16][16][16] = 4096
constexpr int LSS = 5120;                // {sc, sh} pairs    [48][2]      = 96
constexpr int LWV = 5216;                // per-wave regions start
constexpr int PWV = 256 + 48 * WROWS;    // h tile [16][16] + out tile [48][64] = 3328

__global__ __launch_bounds__(WAVES * 32)
void fused_partition_mlp(const float* __restrict__ x,      // (256,64,512)
                         const int*   __restrict__ parts,  // (16,4)
                         const float* __restrict__ W1,     // (16,4,16)
                         const float* __restrict__ b1,     // (16,16)
                         const float* __restrict__ W2,     // (16,16,3)
                         const float* __restrict__ b2,     // (16,3)
                         const float* __restrict__ gamma,  // (16,3)
                         const float* __restrict__ beta,   // (16,3)
                         const float* __restrict__ mean,   // (16,3)
                         const float* __restrict__ var,    // (16,3)
                         float* __restrict__ out)          // (256,48,512)
{
    __shared__ float smem[LWV + WAVES * PWV];
    const int tid = threadIdx.x;

    // ---- one-time prestage: W1 copy ----
    #pragma unroll
    for (int i = 0; i < 4; ++i)
        smem[LW1 + tid + i * 256] = W1[tid + i * 256];
    // ---- one-time prestage: W2 zero-padded from (16,16,3) to [p][k][16] ----
    #pragma unroll
    for (int i = 0; i < 16; ++i) {
        const int e = tid + i * 256;
        const int n = e & 15, k = (e >> 4) & 15, p = e >> 8;
        smem[LW2 + e] = (n < O_) ? W2[(p * H_ + k) * O_ + n] : 0.0f;
    }
    // ---- one-time prestage: BN scale/shift with b2 folded in ----
    if (tid < P_ * O_) {
        const float sc = gamma[tid] * __frsqrt_rn(var[tid] + EPS);
        const float sh = (b2[tid] - mean[tid]) * sc + beta[tid];
        smem[LSS + 2 * tid]     = sc;
        smem[LSS + 2 * tid + 1] = sh;
    }
    __syncthreads();

    const int lane = tid & 31;
    const int wave = tid >> 5;
    float* hl = &smem[LWV + wave * PWV];   // 256 floats:  h tile [16][16]
    float* ol = hl + 256;                  // 3072 floats: out tile [48][64]

    const int job   = blockIdx.x * WAVES + wave;
    const int r0    = job * WROWS;         // 64 | 512 -> never straddles a batch
    const int batch = r0 >> 9;
    const int f0    = r0 & 511;
    const float* xb = x   + (size_t)batch * (NCOL * FEAT) + f0;
    float*       ob = out + (size_t)batch * (P_ * O_ * FEAT) + f0;

    // fp32 16x16x4 WMMA lane roles (wave32):
    //  A: lanes 0-15 -> M=lane, K={0,1}; lanes 16-31 -> M=lane-16, K={2,3}
    //  B: lanes 0-15 -> N=lane, K rows {0,1}; lanes 16-31 -> K rows {2,3}
    //  C/D: vgpr v -> M = v + (lane>=16 ? 8 : 0), N = lane&15
    const int M   = lane & 15;
    const int N   = lane & 15;
    const int kh  = lane >> 4;
    const int klo = kh * 2;
    const int no  = (N < O_) ? N : 0;
    const v8f ZV  = {};                    // WMMA SRC2 inline 0

    #pragma unroll 1
    for (int p = 0; p < P_; ++p) {
        // ---- per-partition parameters (shared across the 4 tiles) ----
        const int2 cc = *(const int2*)&parts[p * IN_ + klo];   // global_load_b64
        v2f w1v;                                               // ds_load_2addr_b32
        w1v.x = smem[LW1 + p * 64 + klo       * 16 + N];
        w1v.y = smem[LW1 + p * 64 + (klo + 1) * 16 + N];
        const float bias1 = b1[p * H_ + N];
        v2f w2v[4];
        #pragma unroll
        for (int ks = 0; ks < 4; ++ks) {                       // 4x ds_load_2addr_b32
            const int kb = ks * 4 + klo;
            w2v[ks].x = smem[LW2 + p * 256 + kb       * 16 + N];
            w2v[ks].y = smem[LW2 + p * 256 + (kb + 1) * 16 + N];
        }
        const float sc = smem[LSS + 2 * (p * O_ + no)];
        const float sh = smem[LSS + 2 * (p * O_ + no) + 1];

        #pragma unroll
        for (int t = 0; t < TPW; ++t) {
            const float* xt = xb + t * 16;
            // ---- layer 1: one WMMA, C = inline 0, bias folded into epilogue ----
            v2f a;
            a.x = xt[cc.x * FEAT + M];
            a.y = xt[cc.y * FEAT + M];
            v8f h = __builtin_amdgcn_wmma_f32_16x16x4_f32(
                false, a, false, w1v, (short)0, ZV, false, false);
            #pragma unroll
            for (int v = 0; v < 8; ++v) {
                float hv = h[v] + bias1;
                hv = fmaxf(hv, SLOPE * hv);                    // leaky ReLU
                hl[(v + kh * 8) * 16 + N] = hv;                // D layout -> [M][K]
            }
            // ---- layer 2: 4 chained K=4 WMMAs, first C = inline 0 ----
            v8f acc = ZV;
            #pragma unroll
            for (int ks = 0; ks < 4; ++ks) {
                const int kb = ks * 4 + klo;                   // even -> 8B aligned
                v2f a2 = *(const v2f*)&hl[M * 16 + kb];        // ds_load_2addr_b64
                acc = __builtin_amdgcn_wmma_f32_16x16x4_f32(
                    false, a2, false, w2v[ks], (short)0, acc, false, false);
            }
            // ---- BN (bias2 folded) + leaky ReLU, stage out tile ----
            if (N < O_) {                                      // reconverges pre-WMMA
                #pragma unroll
                for (int v = 0; v < 8; ++v) {
                    float yv = fmaf(acc[v], sc, sh);
                    yv = fmaxf(yv, SLOPE * yv);
                    ol[(p * O_ + N) * WROWS + t * 16 + v + kh * 8] = yv;
                }
            }
        }
    }

    // ---- coalesced flush: 48x64 floats, float4 per lane (256B segments) ----
    #pragma unroll 1
    for (int j = 0; j < (48 * WROWS) / (32 * 4); ++j) {        // 24 iterations
        const int i = (lane + j * 32) * 4;
        const int k = i >> 6;                                  // channel 0..47
        const int m = i & 63;                                  // feature in wave block
        float4 v4 = *(const float4*)&ol[i];
        *(float4*)&ob[k * FEAT + m] = v4;                      // global_store_b128
    }
}

extern "C" void kernel_launch(void* const* d_in, const int* in_sizes, int n_in,
                              void* d_out, int out_size, void* d_ws, size_t ws_size,
                              hipStream_t stream) {
    const float* x     = (const float*)d_in[0];
    const int*   parts = (const int*)  d_in[1];
    const float* W1    = (const float*)d_in[2];
    const float* b1    = (const float*)d_in[3];
    const float* W2    = (const float*)d_in[4];
    const float* b2    = (const float*)d_in[5];
    const float* gamma = (const float*)d_in[6];
    const float* beta  = (const float*)d_in[7];
    const float* mean  = (const float*)d_in[8];
    const float* var   = (const float*)d_in[9];
    float* out = (float*)d_out;

    dim3 grid(BLOCKS);                 // 256 blocks
    dim3 block(WAVES * 32);            // 256 threads = 8 waves
    hipLaunchKernelGGL(fused_partition_mlp, grid, block, 0, stream,
                       x, parts, W1, b1, W2, b2, gamma, beta, mean, var, out);
}